// ANN_81235011437068
// MI455X (gfx1250) — compile-verified
//
#include <hip/hip_runtime.h>
#include <hip/hip_bf16.h>

#define BATCH   8192
#define N_BANDS 2000
#define TARGET  500
#define HID1    50
#define HID2    10
#define SLOPE   0.01f

typedef __attribute__((ext_vector_type(2))) float v2f;
typedef __attribute__((ext_vector_type(8))) float v8f;

// Workspace layout (float units)
#define WS_INVM 0                         // 2000 floats
#define WS_CP   (N_BANDS)                 // 2000 floats
#define WS_IDX  (2 * N_BANDS)             // 500 ints (stored in float slots)
#define WS_FRAC (2 * N_BANDS + TARGET)    // 500 floats
#define WS_KD   ((size_t)8192)                            // 2000*8192 floats, [i][b]
#define WS_OUTT (WS_KD + (size_t)N_BANDS * BATCH)         // 500*8192 floats, [j][b]

// ---------------------------------------------------------------------------
// Kernel A: batch-independent precompute.
//   - Thomas coefficients cp[i], invm[i] (uniform-h tridiagonal, serial 2000).
//   - Eval-point interval index + fraction from sigmoid(raw_index).
// ---------------------------------------------------------------------------
__global__ void spline_setup_kernel(const float* __restrict__ raw_index,
                                    float* __restrict__ ws) {
    int t = threadIdx.x;
    if (t < TARGET) {
        float r  = raw_index[t];
        float tv = 1.0f / (1.0f + __expf(-r));      // sigmoid, strictly in (0,1)
        float u  = tv * 1999.0f;
        int i    = (int)ceilf(u);                   // searchsorted(times, t, 'left')
        int idx  = i - 1;
        idx = idx < 0 ? 0 : (idx > N_BANDS - 2 ? N_BANDS - 2 : idx);
        ((int*)ws)[WS_IDX + t] = idx;
        ws[WS_FRAC + t] = tv - (float)idx * (1.0f / 1999.0f);
    }
    if (t == 0) {
        const float hr = 1999.0f;
        float m  = 2.0f * hr;                       // diag[0] = 2*hr, lower[0] = 0
        float cp = hr / m;
        ws[WS_INVM + 0] = 1.0f / m;
        ws[WS_CP + 0]   = cp;
        for (int i = 1; i < N_BANDS - 1; ++i) {     // diag = 4*hr interior
            m = 4.0f * hr - hr * cp;
            float inv = 1.0f / m;
            cp = hr * inv;
            ws[WS_INVM + i] = inv;
            ws[WS_CP + i]   = cp;
        }
        m = 2.0f * hr - hr * cp;                    // diag[n-1] = 2*hr
        ws[WS_INVM + N_BANDS - 1] = 1.0f / m;
        ws[WS_CP + N_BANDS - 1]   = 0.0f;           // upper[n-1] = 0
    }
}

// ---------------------------------------------------------------------------
// Kernel B: Thomas solve, one thread per batch row.
//   Forward: stream x[b,:], build rhs on the fly, write dp to kd[i][b]
//   Backward: in-place back-substitution -> kd[i][b] holds spline derivative.
//   kd stored [i][b] so all scratch traffic is lane-coalesced.
// ---------------------------------------------------------------------------
__global__ void thomas_solve_kernel(const float* __restrict__ x,
                                    float* __restrict__ ws) {
    int b = blockIdx.x * blockDim.x + threadIdx.x;
    if (b >= BATCH) return;
    const float hr  = 1999.0f;
    const float hr2 = hr * hr;
    const float* __restrict__ invm = ws + WS_INVM;
    const float* __restrict__ cp   = ws + WS_CP;
    float* __restrict__ kd = ws + WS_KD;
    const float* __restrict__ xr = x + (size_t)b * N_BANDS;

    // forward sweep
    float xp  = xr[0];
    float x1  = xr[1];
    float pds = 3.0f * (x1 - xp) * hr2;
    float dp  = pds * invm[0];                      // rhs[0] = pds[0], lower[0]=0
    kd[(size_t)0 * BATCH + b] = dp;
    xp = x1;
    float pds_prev = pds;
    for (int i = 1; i <= N_BANDS - 2; ++i) {
        float xn = xr[i + 1];
        pds = 3.0f * (xn - xp) * hr2;
        float rhs = pds_prev + pds;
        dp = (rhs - hr * dp) * invm[i];
        kd[(size_t)i * BATCH + b] = dp;
        xp = xn;
        pds_prev = pds;
    }
    dp = (pds_prev - hr * dp) * invm[N_BANDS - 1];  // rhs[n-1] = pds[n-2]
    kd[(size_t)(N_BANDS - 1) * BATCH + b] = dp;

    // backward substitution (in place)
    float kn = dp;                                  // cp[n-1] == 0
    for (int i = N_BANDS - 2; i >= 0; --i) {
        float di = kd[(size_t)i * BATCH + b];
        float k  = di - cp[i] * kn;
        kd[(size_t)i * BATCH + b] = k;
        kn = k;
    }
}

// ---------------------------------------------------------------------------
// Kernel C: fused coefficient construction + spline evaluation.
//   thread = j*BATCH + b (b fastest): coalesced kd gathers and outT writes.
//   outT stored transposed [j][b] to feed the WMMA A-fragments coalesced.
// ---------------------------------------------------------------------------
__global__ void spline_eval_kernel(const float* __restrict__ x,
                                   float* __restrict__ ws) {
    size_t g = (size_t)blockIdx.x * blockDim.x + threadIdx.x;
    if (g >= (size_t)TARGET * BATCH) return;
    int j = (int)(g >> 13);        // / 8192
    int b = (int)(g & 8191);
    const float hr = 1999.0f, hr2 = hr * hr;
    int   idx  = ((const int*)ws)[WS_IDX + j];
    float frac = ws[WS_FRAC + j];
    const float* __restrict__ kd = ws + WS_KD;
    float k0 = kd[(size_t)idx * BATCH + b];
    float k1 = kd[(size_t)(idx + 1) * BATCH + b];
    float a0 = x[(size_t)b * N_BANDS + idx];
    float a1 = x[(size_t)b * N_BANDS + idx + 1];
    float diffs6  = 6.0f * (a1 - a0);
    float two_c   = (diffs6 * hr - 4.0f * k0 - 2.0f * k1) * hr;
    float three_d = (-diffs6 * hr + 3.0f * (k0 + k1)) * hr2;
    float inner = 0.5f * two_c + three_d * frac * (1.0f / 3.0f);
    inner = k0 + inner * frac;                      // b-coeff = kd[b][idx]
    ws[WS_OUTT + (size_t)j * BATCH + b] = a0 + inner * frac;
}

// ---------------------------------------------------------------------------
// Kernel D: MLP head with V_WMMA_F32_16X16X4_F32.
//   Block = 128 threads (4 waves), one 16-row batch tile per block.
//   Wave w computes output columns [16w, 16w+16) of layer 1 (50 cols, padded
//   to 64). K=500 in 125 steps of 4. Column padding handled branchlessly:
//   clamped pointer + hoisted 0/1 mask keeps the inner loop a pure
//   load-clause + wmma body (no per-iteration exec-mask manipulation).
// ---------------------------------------------------------------------------
__global__ __launch_bounds__(128) void mlp_wmma_kernel(
    const float* __restrict__ outT,   // [TARGET][BATCH]
    const float* __restrict__ W1, const float* __restrict__ b1,
    const float* __restrict__ W2, const float* __restrict__ b2,
    const float* __restrict__ W3, const float* __restrict__ b3,
    float* __restrict__ out) {
    __shared__ float h1[16][65];                    // padded: kills bank conflicts
    int tid  = threadIdx.x;
    int lane = tid & 31;
    int wave = tid >> 5;                            // 0..3
    int b0   = blockIdx.x * 16;
    int n0   = wave * 16;
    int mrow = lane & 15;                           // A: M index / B: N index
    int ksel = (lane >> 4) * 2;                     // lanes 0-15 -> K=0,1; 16-31 -> K=2,3
    int ncol = n0 + mrow;

    // Hoisted branchless padding for columns >= 50: always-valid address,
    // loaded value multiplied by 0.
    const bool  valid = (ncol < HID1);
    const float wmask = valid ? 1.0f : 0.0f;
    const float* __restrict__ Wcol = W1 + (valid ? ncol : 0);
    const float* __restrict__ Acol = outT + b0 + mrow;

    v8f acc = {};
    for (int k0 = 0; k0 < TARGET; k0 += 4) {
        int ka = k0 + ksel;
        v2f af, bf;
        // A fragment: A[m][k] = outT[k][b0+m] (16 consecutive b -> coalesced)
        af.x = Acol[(size_t)ka * BATCH];
        af.y = Acol[(size_t)(ka + 1) * BATCH];
        // B fragment: B[k][n] = W1[k][n], zero-masked beyond col 49
        bf.x = Wcol[(size_t)ka * HID1] * wmask;
        bf.y = Wcol[(size_t)(ka + 1) * HID1] * wmask;
        acc = __builtin_amdgcn_wmma_f32_16x16x4_f32(
            /*neg_a=*/false, af, /*neg_b=*/false, bf,
            /*c_mod=*/(short)0, acc, /*reuse_a=*/false, /*reuse_b=*/false);
    }

    // D layout: lanes 0-15 hold rows v (VGPR v), lanes 16-31 hold rows v+8
    int   rbase = (lane >> 4) * 8;
    float bias  = valid ? b1[ncol] : 0.0f;
#pragma unroll
    for (int v = 0; v < 8; ++v) {
        float val = acc[v] + bias;
        val = (val >= 0.0f) ? val : SLOPE * val;    // LeakyReLU
        h1[rbase + v][ncol] = valid ? val : 0.0f;
    }
    __syncthreads();

    if (tid < 16) {                                 // one thread per batch row
        float acc3 = b3[0];
        for (int j = 0; j < HID2; ++j) {
            float s = b2[j];
            for (int k = 0; k < HID1; ++k) s += h1[tid][k] * W2[k * HID2 + j];
            s = (s >= 0.0f) ? s : SLOPE * s;
            acc3 += s * W3[j];
        }
        out[b0 + tid] = acc3;
    }
}

// ---------------------------------------------------------------------------
extern "C" void kernel_launch(void* const* d_in, const int* in_sizes, int n_in,
                              void* d_out, int out_size, void* d_ws, size_t ws_size,
                              hipStream_t stream) {
    const float* x         = (const float*)d_in[0];
    const float* raw_index = (const float*)d_in[1];
    const float* W1        = (const float*)d_in[2];
    const float* b1        = (const float*)d_in[3];
    const float* W2        = (const float*)d_in[4];
    const float* b2        = (const float*)d_in[5];
    const float* W3        = (const float*)d_in[6];
    const float* b3        = (const float*)d_in[7];
    float* ws  = (float*)d_ws;
    float* out = (float*)d_out;

    spline_setup_kernel<<<1, 512, 0, stream>>>(raw_index, ws);
    thomas_solve_kernel<<<BATCH / 256, 256, 0, stream>>>(x, ws);
    size_t total = (size_t)TARGET * BATCH;
    spline_eval_kernel<<<(unsigned)((total + 255) / 256), 256, 0, stream>>>(x, ws);
    mlp_wmma_kernel<<<BATCH / 16, 128, 0, stream>>>(ws + WS_OUTT, W1, b1, W2, b2,
                                                    W3, b3, out);
}